// LGHGCLNetV2_56057913147943
// MI455X (gfx1250) — compile-verified
//
#include <hip/hip_runtime.h>
#include <hip/hip_bf16.h>
#include <math.h>

// ---- problem constants (match reference) ----
#define NNODES 100000
#define NEDGES 1600000
#define DIN    64
#define MHID   256
#define GHID   128
#define OHID   128
#define RREL   3
#define TAU_INV 10.0f
#define EPSBN   1e-5f

typedef float v2f __attribute__((ext_vector_type(2)));
typedef float v8f __attribute__((ext_vector_type(8)));

// =====================================================================
// fp32 WMMA GEMM: D[M,Nc] = A[M,K] @ B[K,Nc] (+Cadd) (+bias)
// Optional batch over blockIdx.z applied to B and D (for per-relation xr).
// Each wave computes a 16x64 output strip via 4x V_WMMA_F32_16X16X4_F32.
// Layouts per CDNA5 ISA 7.12.2:
//   A 16x4 f32 : lanes 0-15 -> M=lane (K=0 in v0, K=1 in v1); lanes 16-31 -> K=2,3
//   B  4x16    : v0 lanes0-15 = B[K=0][n], lanes16-31 = B[K=2][n]; v1 = K=1/K=3
//   C/D        : vj lanes0-15 = (M=j, N=n); lanes16-31 = (M=j+8, N=n)
// =====================================================================
__global__ __launch_bounds__(128) void wmma_gemm_f32(
    const float* __restrict__ A, const float* __restrict__ B,
    const float* __restrict__ Cadd, const float* __restrict__ bias,
    float* __restrict__ Dst, int M, int K, int Nc)
{
    const int lane = threadIdx.x & 31;
    const int wave = threadIdx.x >> 5;
    const int half = lane >> 4;      // 0: K=0,1 / 1: K=2,3 (A); row+0 / row+8 (C)
    const int mh   = lane & 15;      // M index (A) / N index (B,C)
    const long r0  = ((long)blockIdx.y * 4 + wave) * 16;
    if (r0 >= M) return;             // wave-uniform; N multiple of 16 so tiles are all-or-nothing
    const int  c0   = blockIdx.x * 64;
    const long bOff = (long)blockIdx.z * K * Nc;
    const long dOff = (long)blockIdx.z * (long)M * Nc;
    const float* Ar = A + (r0 + mh) * (long)K;

    v8f acc[4] = {v8f{}, v8f{}, v8f{}, v8f{}};

    for (int k0 = 0; k0 < K; k0 += 4) {
        v2f a;
        a.x = Ar[k0 + 2 * half + 0];
        a.y = Ar[k0 + 2 * half + 1];
        const float* Brow0 = B + bOff + (long)(k0 + 2 * half) * Nc + c0 + mh;
        const float* Brow1 = Brow0 + Nc;
        v2f b0; b0.x = Brow0[ 0]; b0.y = Brow1[ 0];
        v2f b1; b1.x = Brow0[16]; b1.y = Brow1[16];
        v2f b2; b2.x = Brow0[32]; b2.y = Brow1[32];
        v2f b3; b3.x = Brow0[48]; b3.y = Brow1[48];
        acc[0] = __builtin_amdgcn_wmma_f32_16x16x4_f32(false, a, false, b0, (short)0, acc[0], false, false);
        acc[1] = __builtin_amdgcn_wmma_f32_16x16x4_f32(false, a, false, b1, (short)0, acc[1], false, false);
        acc[2] = __builtin_amdgcn_wmma_f32_16x16x4_f32(false, a, false, b2, (short)0, acc[2], false, false);
        acc[3] = __builtin_amdgcn_wmma_f32_16x16x4_f32(false, a, false, b3, (short)0, acc[3], false, false);
    }

#pragma unroll
    for (int t = 0; t < 4; ++t) {
        const int col = c0 + t * 16 + mh;
        const float bv = bias ? bias[col] : 0.0f;
#pragma unroll
        for (int j = 0; j < 8; ++j) {
            const long row = r0 + j + 8 * half;
            const long idx = row * Nc + col;
            float v = acc[t][j] + bv;
            if (Cadd) v += Cadd[idx];
            Dst[dOff + idx] = v;
        }
    }
}

// ---------------- elementwise / init helpers ----------------
__global__ void fill_f32(float* __restrict__ p, long n, float v)
{
    long i = (long)blockIdx.x * blockDim.x + threadIdx.x;
    long s = (long)gridDim.x * blockDim.x;
    for (; i < n; i += s) p[i] = v;
}

__global__ void copy_f32(const float* __restrict__ src, float* __restrict__ dst, long n)
{
    long i = (long)blockIdx.x * blockDim.x + threadIdx.x;
    long s = (long)gridDim.x * blockDim.x;
    for (; i < n; i += s) dst[i] = src[i];
}

// ---------------- batch-norm (column mean/var over N rows) ----------------
#define STAT_ROWS 256
__global__ __launch_bounds__(256) void col_stats(const float* __restrict__ Z,
    int M, int C, float* __restrict__ sums, float* __restrict__ sqs)
{
    __shared__ float ls[256];
    __shared__ float lq[256];
    if ((int)threadIdx.x < C) { ls[threadIdx.x] = 0.f; lq[threadIdx.x] = 0.f; }
    __syncthreads();
    const long base  = (long)blockIdx.x * STAT_ROWS;
    const long total = (long)STAT_ROWS * C;
    for (long i = threadIdx.x; i < total; i += blockDim.x) {
        const long row = base + i / C;
        const int  col = (int)(i % C);
        if (row < M) {
            float v = Z[row * C + col];
            atomicAdd(&ls[col], v);       // ds_add_f32
            atomicAdd(&lq[col], v * v);
        }
    }
    __syncthreads();
    if ((int)threadIdx.x < C) {
        atomicAdd(&sums[threadIdx.x], ls[threadIdx.x]);
        atomicAdd(&sqs [threadIdx.x], lq[threadIdx.x]);
    }
}

__global__ void bn_finalize(const float* __restrict__ sums, const float* __restrict__ sqs,
    const float* __restrict__ g, const float* __restrict__ b, int M, int C,
    float* __restrict__ scsh)
{
    int c = blockIdx.x * blockDim.x + threadIdx.x;
    if (c < C) {
        float mean = sums[c] / (float)M;
        float var  = sqs[c] / (float)M - mean * mean;
        float sc   = g[c] * rsqrtf(var + EPSBN);
        scsh[c]     = sc;
        scsh[C + c] = b[c] - mean * sc;
    }
}

__global__ void bn_apply_relu(const float* __restrict__ Z, const float* __restrict__ scsh,
    float* __restrict__ Out, long total, int C)
{
    long i = (long)blockIdx.x * blockDim.x + threadIdx.x;
    long s = (long)gridDim.x * blockDim.x;
    for (; i < total; i += s) {
        int c = (int)(i % C);
        float v = Z[i] * scsh[c] + scsh[C + c];
        Out[i] = v > 0.f ? v : 0.f;
    }
}

// ---------------- ASDA: edge softmax over incoming edges per dst ----------------
__device__ __forceinline__ void atomicMaxF32(float* addr, float val)
{
    if (val >= 0.0f) atomicMax((int*)addr, __float_as_int(val));
    else             atomicMin((unsigned int*)addr, __float_as_uint(val));
}

__global__ void edge_smax(const float* __restrict__ hsd, const int* __restrict__ src,
    const int* __restrict__ dst, float* __restrict__ smax, int E)
{
    int e = blockIdx.x * blockDim.x + threadIdx.x;
    if (e < E) {
        float s = (hsd[src[e]] - hsd[dst[e]]) * TAU_INV;
        atomicMaxF32(&smax[dst[e]], s);
    }
}

__global__ void edge_expsum(const float* __restrict__ hsd, const int* __restrict__ src,
    const int* __restrict__ dst, const float* __restrict__ smax,
    float* __restrict__ expv, float* __restrict__ den, int E)
{
    int e = blockIdx.x * blockDim.x + threadIdx.x;
    if (e < E) {
        int d = dst[e];
        float s = (hsd[src[e]] - hsd[d]) * TAU_INV;
        float x = __expf(s - smax[d]);
        expv[e] = x;
        atomicAdd(&den[d], x);
    }
}

// wave-per-edge scatter: hin[dst] += alpha * x[src]  (DIN = 64 -> 2 floats/lane)
__global__ __launch_bounds__(256) void edge_scatter_asda(const float* __restrict__ x,
    const int* __restrict__ src, const int* __restrict__ dst,
    const float* __restrict__ expv, const float* __restrict__ den,
    float* __restrict__ hin, int E)
{
    int wid  = (int)((blockIdx.x * blockDim.x + threadIdx.x) >> 5);
    int lane = threadIdx.x & 31;
    if (wid >= E) return;
    int s = src[wid], d = dst[wid];
    float alpha = expv[wid] / fmaxf(den[d], 1e-9f);
    float v0 = x[(long)s * DIN + lane];
    float v1 = x[(long)s * DIN + 32 + lane];
    atomicAdd(&hin[(long)d * DIN + lane],       alpha * v0);
    atomicAdd(&hin[(long)d * DIN + 32 + lane],  alpha * v1);
}

// ---------------- RGCN edge scatter ----------------
__global__ void edge_count(const int* __restrict__ dst, const int* __restrict__ et,
    float* __restrict__ cnt, int E)
{
    int e = blockIdx.x * blockDim.x + threadIdx.x;
    if (e < E) atomicAdd(&cnt[(long)et[e] * NNODES + dst[e]], 1.0f);
}

// wave-per-edge: agg[dst] += xr[etype, src] / max(cnt[etype,dst], 1)
__global__ __launch_bounds__(256) void edge_scatter_rgcn(const float* __restrict__ xr,
    const int* __restrict__ src, const int* __restrict__ dst, const int* __restrict__ et,
    const float* __restrict__ cnt, float* __restrict__ agg, int E, int H)
{
    int wid  = (int)((blockIdx.x * blockDim.x + threadIdx.x) >> 5);
    int lane = threadIdx.x & 31;
    if (wid >= E) return;
    int r = et[wid], s = src[wid], d = dst[wid];
    float w = 1.0f / fmaxf(cnt[(long)r * NNODES + d], 1.0f);
    const float* m = xr + ((long)r * NNODES + s) * H;
    float* a = agg + (long)d * H;
    for (int j = lane; j < H; j += 32)
        atomicAdd(&a[j], m[j] * w);
}

// ---------------- output head: logits = [zmlp | zgnn] @ out_w + out_b ----------------
__global__ __launch_bounds__(256) void head_kernel(const float* __restrict__ zmlp,
    const float* __restrict__ zgnn, const float* __restrict__ ow,
    const float* __restrict__ ob, float* __restrict__ out, int Nn)
{
    int wid  = (int)((blockIdx.x * blockDim.x + threadIdx.x) >> 5);
    int lane = threadIdx.x & 31;
    if (wid >= Nn) return;
    const float* zm = zmlp + (long)wid * OHID;
    const float* zg = zgnn + (long)wid * OHID;
    float acc = 0.f;
#pragma unroll
    for (int j0 = 0; j0 < OHID; j0 += 32) {
        int j = j0 + lane;
        acc += zm[j] * ow[j];
        acc += zg[j] * ow[OHID + j];
    }
#pragma unroll
    for (int off = 16; off > 0; off >>= 1) acc += __shfl_xor(acc, off, 32);
    if (lane == 0) out[wid] = acc + ob[0];
}

// =====================================================================
// Host orchestration
// =====================================================================
static inline int cdiv(long a, long b) { return (int)((a + b - 1) / b); }

extern "C" void kernel_launch(void* const* d_in, const int* in_sizes, int n_in,
                              void* d_out, int out_size, void* d_ws, size_t ws_size,
                              hipStream_t stream)
{
    (void)in_sizes; (void)n_in; (void)out_size; (void)ws_size;
    const float* x       = (const float*)d_in[0];
    const float* hsd     = (const float*)d_in[1];
    const int*   ei      = (const int*)  d_in[2];   // (2, E)
    const int*   et      = (const int*)  d_in[3];
    const float* mlp_w1  = (const float*)d_in[4];
    const float* mlp_b1  = (const float*)d_in[5];
    const float* mlp_g1  = (const float*)d_in[6];
    const float* mlp_be1 = (const float*)d_in[7];
    const float* mlp_w2  = (const float*)d_in[8];
    const float* mlp_b2  = (const float*)d_in[9];
    const float* mlp_g2  = (const float*)d_in[10];
    const float* mlp_be2 = (const float*)d_in[11];
    const float* rg1_wrel  = (const float*)d_in[12];
    const float* rg1_wroot = (const float*)d_in[13];
    const float* rg1_b     = (const float*)d_in[14];
    const float* bn1_g     = (const float*)d_in[15];
    const float* bn1_b     = (const float*)d_in[16];
    const float* rg2_wrel  = (const float*)d_in[17];
    const float* rg2_wroot = (const float*)d_in[18];
    const float* rg2_b     = (const float*)d_in[19];
    const float* bn2_g     = (const float*)d_in[20];
    const float* bn2_b     = (const float*)d_in[21];
    const float* out_w     = (const float*)d_in[22];
    const float* out_b     = (const float*)d_in[23];
    // d_in[24..27] (proj head) unused by the reference output.

    const int* srcp = ei;
    const int* dstp = ei + NEDGES;
    float* out = (float*)d_out;
    float* ws  = (float*)d_ws;

    // workspace layout (floats). O_BIG aliases MLP hidden h1 (N*256) and xr (R*N*128):
    const size_t O_BIG  = 0;
    const size_t O_ZMLP = O_BIG  + (size_t)RREL * NNODES * 128;   // 38.4M
    const size_t O_HIN  = O_ZMLP + (size_t)NNODES * OHID;
    const size_t O_AGG  = O_HIN  + (size_t)NNODES * DIN;
    const size_t O_Z1   = O_AGG  + (size_t)NNODES * 128;
    const size_t O_Z2   = O_Z1   + (size_t)NNODES * 128;
    const size_t O_EXPV = O_Z2   + (size_t)NNODES * 128;
    const size_t O_SMAX = O_EXPV + (size_t)NEDGES;
    const size_t O_DEN  = O_SMAX + (size_t)NNODES;
    const size_t O_CNT  = O_DEN  + (size_t)NNODES;
    const size_t O_SUMS = O_CNT  + (size_t)RREL * NNODES;
    const size_t O_SQS  = O_SUMS + 256;
    const size_t O_SCSH = O_SQS  + 256;

    const int rowBlocks  = cdiv(NNODES, 64);
    const int statBlocks = cdiv(NNODES, STAT_ROWS);
    const int edgeBlocks = cdiv(NEDGES, 256);
    const int edgeWaveBl = cdiv((long)NEDGES * 32, 256);
    const int nodeWaveBl = cdiv((long)NNODES * 32, 256);

#define FILL(ptr, n, v) fill_f32<<<cdiv((long)(n), 256 * 4), 256, 0, stream>>>((ptr), (long)(n), (v))
#define BN_PASS(Z, C, g, b)                                                                         \
    do {                                                                                            \
        FILL(ws + O_SUMS, 512, 0.0f);                                                               \
        col_stats<<<statBlocks, 256, 0, stream>>>((Z), NNODES, (C), ws + O_SUMS, ws + O_SQS);       \
        bn_finalize<<<1, 256, 0, stream>>>(ws + O_SUMS, ws + O_SQS, (g), (b), NNODES, (C),          \
                                           ws + O_SCSH);                                            \
        bn_apply_relu<<<cdiv((long)NNODES * (C), 256 * 4), 256, 0, stream>>>(                       \
            (Z), ws + O_SCSH, (Z), (long)NNODES * (C), (C));                                        \
    } while (0)

    // ---------- MLP branch ----------
    // h1 = x @ mlp_w1 + b1 ; relu(bn(h1)) in place
    wmma_gemm_f32<<<dim3(MHID / 64, rowBlocks, 1), 128, 0, stream>>>(
        x, mlp_w1, nullptr, mlp_b1, ws + O_BIG, NNODES, DIN, MHID);
    BN_PASS(ws + O_BIG, MHID, mlp_g1, mlp_be1);
    // z_mlp = h1 @ mlp_w2 + b2 ; relu(bn(.)) in place
    wmma_gemm_f32<<<dim3(OHID / 64, rowBlocks, 1), 128, 0, stream>>>(
        ws + O_BIG, mlp_w2, nullptr, mlp_b2, ws + O_ZMLP, NNODES, MHID, OHID);
    BN_PASS(ws + O_ZMLP, OHID, mlp_g2, mlp_be2);

    // ---------- ASDA ----------
    FILL(ws + O_SMAX, NNODES, -INFINITY);
    FILL(ws + O_DEN,  NNODES, 0.0f);
    copy_f32<<<cdiv((long)NNODES * DIN, 256 * 4), 256, 0, stream>>>(x, ws + O_HIN, (long)NNODES * DIN);
    edge_smax<<<edgeBlocks, 256, 0, stream>>>(hsd, srcp, dstp, ws + O_SMAX, NEDGES);
    edge_expsum<<<edgeBlocks, 256, 0, stream>>>(hsd, srcp, dstp, ws + O_SMAX,
                                                ws + O_EXPV, ws + O_DEN, NEDGES);
    edge_scatter_asda<<<edgeWaveBl, 256, 0, stream>>>(x, srcp, dstp, ws + O_EXPV,
                                                      ws + O_DEN, ws + O_HIN, NEDGES);

    // per-(relation,dst) edge counts (identical for both RGCN layers)
    FILL(ws + O_CNT, (size_t)RREL * NNODES, 0.0f);
    edge_count<<<edgeBlocks, 256, 0, stream>>>(dstp, et, ws + O_CNT, NEDGES);

    // ---------- RGCN layer 1 ----------
    // xr[r] = hin @ rg1_wrel[r]   (batched over blockIdx.z)
    wmma_gemm_f32<<<dim3(GHID / 64, rowBlocks, RREL), 128, 0, stream>>>(
        ws + O_HIN, rg1_wrel, nullptr, nullptr, ws + O_BIG, NNODES, DIN, GHID);
    FILL(ws + O_AGG, (size_t)NNODES * GHID, 0.0f);
    edge_scatter_rgcn<<<edgeWaveBl, 256, 0, stream>>>(ws + O_BIG, srcp, dstp, et,
                                                      ws + O_CNT, ws + O_AGG, NEDGES, GHID);
    // z1 = hin @ rg1_wroot + agg + b ; relu(bn(.)) in place
    wmma_gemm_f32<<<dim3(GHID / 64, rowBlocks, 1), 128, 0, stream>>>(
        ws + O_HIN, rg1_wroot, ws + O_AGG, rg1_b, ws + O_Z1, NNODES, DIN, GHID);
    BN_PASS(ws + O_Z1, GHID, bn1_g, bn1_b);

    // ---------- RGCN layer 2 ----------
    wmma_gemm_f32<<<dim3(OHID / 64, rowBlocks, RREL), 128, 0, stream>>>(
        ws + O_Z1, rg2_wrel, nullptr, nullptr, ws + O_BIG, NNODES, GHID, OHID);
    FILL(ws + O_AGG, (size_t)NNODES * OHID, 0.0f);
    edge_scatter_rgcn<<<edgeWaveBl, 256, 0, stream>>>(ws + O_BIG, srcp, dstp, et,
                                                      ws + O_CNT, ws + O_AGG, NEDGES, OHID);
    wmma_gemm_f32<<<dim3(OHID / 64, rowBlocks, 1), 128, 0, stream>>>(
        ws + O_Z1, rg2_wroot, ws + O_AGG, rg2_b, ws + O_Z2, NNODES, GHID, OHID);
    BN_PASS(ws + O_Z2, OHID, bn2_g, bn2_b);

    // ---------- fused head ----------
    head_kernel<<<nodeWaveBl, 256, 0, stream>>>(ws + O_ZMLP, ws + O_Z2, out_w, out_b,
                                                out, NNODES);
#undef BN_PASS
#undef FILL
}